// RBFNN_34660386078866
// MI455X (gfx1250) — compile-verified
//
#include <hip/hip_runtime.h>
#include <math.h>

#define NODES   100000
#define DEG     16
#define IN_DIM  128
#define OUT_DIM 32
#define EPS     1e-8f

typedef __attribute__((ext_vector_type(2))) float v2f;
typedef __attribute__((ext_vector_type(8))) float v8f;

// -------------------------------------------------------------------------
// Kernel 1: z = h @ W.T via V_WMMA_F32_16X16X4_F32.
// One wave per 16-row tile; two 16x16 f32 accumulators cover all 32 cols.
// A (16x4 f32): lane<16 -> {K=k,k+1}, lane>=16 -> {K=k+2,k+3} for row M=lane&15.
// B (4x16 f32): mirrored layout; B[k][n] = W[n][k], so each lane loads a
// contiguous float2 from row (n0 + lane&15) of W.
// -------------------------------------------------------------------------
__global__ void __launch_bounds__(128)
gat_gemm_wmma(const float* __restrict__ h, const float* __restrict__ W,
              float* __restrict__ z) {
    const int wave = blockIdx.x * (blockDim.x >> 5) + (threadIdx.x >> 5);
    if (wave >= NODES / 16) return;          // wave-uniform: EXEC stays all-1s
    const int lane = threadIdx.x & 31;
    const int m    = lane & 15;
    const int hlf  = lane >> 4;
    const int row0 = wave * 16;

    const float* ha = h + (size_t)(row0 + m) * IN_DIM + 2 * hlf;
    const float* wb0 = W + (size_t)m        * IN_DIM + 2 * hlf;   // cols  0..15
    const float* wb1 = W + (size_t)(16 + m) * IN_DIM + 2 * hlf;   // cols 16..31

    v8f c0 = {};
    v8f c1 = {};
#pragma unroll 4
    for (int k = 0; k < IN_DIM; k += 4) {
        v2f a  = *(const v2f*)(ha  + k);
        v2f b0 = *(const v2f*)(wb0 + k);
        v2f b1 = *(const v2f*)(wb1 + k);
        c0 = __builtin_amdgcn_wmma_f32_16x16x4_f32(false, a, false, b0,
                                                   (short)0, c0, false, false);
        c1 = __builtin_amdgcn_wmma_f32_16x16x4_f32(false, a, false, b1,
                                                   (short)0, c1, false, false);
    }

    // D layout: VGPR v, lane<16 -> (M=v, N=lane); lane>=16 -> (M=v+8, N=lane-16)
    const int r = row0 + hlf * 8;
#pragma unroll
    for (int v = 0; v < 8; ++v) {
        z[(size_t)(r + v) * OUT_DIM + m]      = c0[v];
        z[(size_t)(r + v) * OUT_DIM + 16 + m] = c1[v];
    }
}

// -------------------------------------------------------------------------
// Kernel 2: per-node clamped L2 norm of z rows.
// -------------------------------------------------------------------------
__global__ void __launch_bounds__(256)
gat_norms(const float* __restrict__ z, float* __restrict__ nrm) {
    const int n = blockIdx.x * blockDim.x + threadIdx.x;
    if (n >= NODES) return;
    const float4* zp = (const float4*)(z + (size_t)n * OUT_DIM);
    float s = 0.f;
#pragma unroll
    for (int i = 0; i < OUT_DIM / 4; ++i) {
        float4 q = zp[i];
        s += q.x * q.x + q.y * q.y + q.z * q.z + q.w * q.w;
    }
    nrm[n] = fmaxf(sqrtf(s), EPS);
}

// -------------------------------------------------------------------------
// Kernel 3: one wave32 per node. Edges of node n are [16n, 16n+16) since
// dst = arange(E)//DEG. Lane l handles edge e=l&15, half h=l>>4 of the dot.
// Softmax over the 16 edges via butterfly shfl_xor; then transposed
// aggregation: lane d accumulates sum_e alpha_e * z[src_e][d] (coalesced).
// -------------------------------------------------------------------------
__global__ void __launch_bounds__(256)
gat_edges(const float* __restrict__ z, const float* __restrict__ nrm,
          const int* __restrict__ src, const float* __restrict__ betaPtr,
          float* __restrict__ out) {
    const int wavesPerBlk = blockDim.x >> 5;
    const int n = blockIdx.x * wavesPerBlk + ((int)threadIdx.x >> 5);
    if (n >= NODES) return;                  // wave-uniform
    const int lane = threadIdx.x & 31;
    const int e    = lane & 15;
    const int hlf  = lane >> 4;
    const float beta = betaPtr[0];

    const int s = src[n * DEG + e];

    // ---- cosine similarity for edge e ----
    const float* zs = z + (size_t)s * OUT_DIM + hlf * 16;
    const float* zd = z + (size_t)n * OUT_DIM + hlf * 16;
    float dot = 0.f;
#pragma unroll
    for (int i = 0; i < 16; ++i) dot = fmaf(zs[i], zd[i], dot);
    dot += __shfl_xor(dot, 16, 32);          // combine halves: all 32 lanes full dot

    const float cosv = dot / (nrm[s] * nrm[n]);
    const float esc  = -beta * (1.0f - cosv);

    // ---- stable softmax over 16 edges (values duplicated across halves) ----
    float mx = esc;
    mx = fmaxf(mx, __shfl_xor(mx, 1, 32));
    mx = fmaxf(mx, __shfl_xor(mx, 2, 32));
    mx = fmaxf(mx, __shfl_xor(mx, 4, 32));
    mx = fmaxf(mx, __shfl_xor(mx, 8, 32));
    float ex = __expf(esc - mx);
    float dn = ex;
    dn += __shfl_xor(dn, 1, 32);
    dn += __shfl_xor(dn, 2, 32);
    dn += __shfl_xor(dn, 4, 32);
    dn += __shfl_xor(dn, 8, 32);
    const float alpha = ex / dn;

    // ---- aggregation: lane d in [0,32) owns output dim d ----
    float acc = 0.f;
#pragma unroll
    for (int j = 0; j < DEG; ++j) {
        const float a  = __shfl(alpha, j, 32);   // edge j's weight (lane j)
        const int   sj = __shfl(s,     j, 32);   // edge j's source node
        acc = fmaf(a, z[(size_t)sj * OUT_DIM + lane], acc);
    }
    out[(size_t)n * OUT_DIM + lane] = acc;
}

// -------------------------------------------------------------------------
extern "C" void kernel_launch(void* const* d_in, const int* in_sizes, int n_in,
                              void* d_out, int out_size, void* d_ws, size_t ws_size,
                              hipStream_t stream) {
    const float* h    = (const float*)d_in[0];   // [N, 128]
    const float* W    = (const float*)d_in[1];   // [32, 128]
    const float* beta = (const float*)d_in[2];   // [1]
    const int*   src  = (const int*)d_in[3];     // [E]
    // d_in[4] (dst) is structurally arange(E)//DEG -> edges of node n contiguous
    float* out = (float*)d_out;

    float* z   = (float*)d_ws;                   // N*32 floats = 12.8 MB
    float* nrm = z + (size_t)NODES * OUT_DIM;    // N floats

    // GEMM: 6250 row-tiles, 4 waves (128 thr) per block
    const int tiles = NODES / 16;                            // 6250
    const int wpb   = 4;
    gat_gemm_wmma<<<dim3((tiles + wpb - 1) / wpb), dim3(wpb * 32), 0, stream>>>(h, W, z);

    gat_norms<<<dim3((NODES + 255) / 256), dim3(256), 0, stream>>>(z, nrm);

    // Edge/softmax/aggregate: 8 waves (256 thr) per block, one wave per node
    gat_edges<<<dim3((NODES + 7) / 8), dim3(256), 0, stream>>>(z, nrm, src, beta, out);
}